// GCN_5506148074002
// MI455X (gfx1250) — compile-verified
//
#include <hip/hip_runtime.h>

#define N_NODES 100000
#define N_EDGES 600000
#define F       128
#define N_CLS   40
#define NPAD    48

typedef float v2f __attribute__((ext_vector_type(2)));
typedef float v8f __attribute__((ext_vector_type(8)));

// ---------------- degree / normalization ----------------
__global__ void gcn_init_deg(float* deg) {
    int i = blockIdx.x * blockDim.x + threadIdx.x;
    if (i < N_NODES) deg[i] = 1.0f;   // self-loop contributes 1
}

__global__ void gcn_count_deg(const int* __restrict__ ei, float* deg) {
    int i = blockIdx.x * blockDim.x + threadIdx.x;
    if (i < N_EDGES) atomicAdd(&deg[ei[N_EDGES + i]], 1.0f);  // dst row
}

__global__ void gcn_make_dis(float* deg) {
    int i = blockIdx.x * blockDim.x + threadIdx.x;
    if (i < N_NODES) deg[i] = rsqrtf(deg[i]);   // deg >= 1 always
}

// ---------------- zero a feature buffer (float4) ----------------
__global__ void gcn_zero(float4* p, int n4) {
    int i = blockIdx.x * blockDim.x + threadIdx.x;
    int stride = gridDim.x * blockDim.x;
    float4 z = {0.f, 0.f, 0.f, 0.f};
    for (; i < n4; i += stride) p[i] = z;
}

// ---------------- pad linW (128x40) -> Wp (128x48), linb -> bp(48) ----------------
__global__ void gcn_pad_w(const float* __restrict__ linW, const float* __restrict__ linb,
                          float* __restrict__ Wp, float* __restrict__ bp) {
    int i = blockIdx.x * blockDim.x + threadIdx.x;
    if (i < F * NPAD) {
        int r = i / NPAD, c = i - r * NPAD;
        Wp[i] = (c < N_CLS) ? linW[r * N_CLS + c] : 0.0f;
    }
    if (i < NPAD) bp[i] = (i < N_CLS) ? linb[i] : 0.0f;
}

// ---------------- T = X @ W  (M=100000, N=128, K=128), fp32 WMMA ----------------
// 256 threads = 8 waves; block owns 16 rows x 128 cols; wave w owns cols [16w,16w+16)
__global__ __launch_bounds__(256) void gcn_gemm128(const float* __restrict__ X,
                                                   const float* __restrict__ W,
                                                   float* __restrict__ T) {
    __shared__ float As[16 * 132];           // padded stride kills LDS bank conflicts
    const int t    = threadIdx.x;
    const int row0 = blockIdx.x * 16;

    #pragma unroll
    for (int j = 0; j < 8; ++j) {            // 2048 floats, coalesced
        int idx = t * 8 + j;
        int r = idx >> 7, c = idx & 127;
        As[r * 132 + c] = X[(size_t)(row0 + r) * F + c];
    }
    __syncthreads();

    const int lane = t & 31;
    const int half = lane >> 4;              // 0: lanes 0-15, 1: lanes 16-31
    const int m    = lane & 15;
    const int n0   = (t >> 5) * 16;

    v8f acc = {};
    #pragma unroll 8
    for (int k = 0; k < F; k += 4) {
        v2f a, b;
        // A 16x4: VGPR0 = K=k+2*half, VGPR1 = K=k+2*half+1  (M = m)
        a.x = As[m * 132 + k + 2 * half];
        a.y = As[m * 132 + k + 2 * half + 1];
        // B 4x16: row K striped across lanes within a VGPR
        b.x = W[(size_t)(k + 2 * half) * F + n0 + m];
        b.y = W[(size_t)(k + 2 * half + 1) * F + n0 + m];
        acc = __builtin_amdgcn_wmma_f32_16x16x4_f32(false, a, false, b,
                                                    (short)0, acc, false, false);
    }

    #pragma unroll
    for (int r = 0; r < 8; ++r)              // VGPR r: M = r (+8 for upper half)
        T[(size_t)(row0 + r + 8 * half) * F + n0 + m] = acc[r];
}

// ---------------- edge scatter: agg[dst] += T[src] * dis[src]*dis[dst] ----------------
__global__ __launch_bounds__(256) void gcn_scatter(const int* __restrict__ ei,
                                                   const float* __restrict__ T,
                                                   const float* __restrict__ dis,
                                                   float* agg) {
    int wid   = (blockIdx.x * blockDim.x + threadIdx.x) >> 5;
    int lane  = threadIdx.x & 31;
    int nwave = (gridDim.x * blockDim.x) >> 5;
    for (int e = wid; e < N_EDGES; e += nwave) {
        int s = ei[e];
        int d = ei[N_EDGES + e];
        float nrm = dis[s] * dis[d];
        const float4 hv = *(const float4*)(T + (size_t)s * F + lane * 4);
        float* ap = agg + (size_t)d * F + lane * 4;
        atomicAdd(ap + 0, hv.x * nrm);
        atomicAdd(ap + 1, hv.y * nrm);
        atomicAdd(ap + 2, hv.z * nrm);
        atomicAdd(ap + 3, hv.w * nrm);
    }
}

// ---------------- finalize (in place): agg = relu(agg + T*dis^2 + b) ----------------
__global__ void gcn_finalize(float4* agg, const float4* __restrict__ T,
                             const float* __restrict__ dis, const float* __restrict__ b) {
    int i = blockIdx.x * blockDim.x + threadIdx.x;
    int stride = gridDim.x * blockDim.x;
    const int n4 = N_NODES * (F / 4);
    for (; i < n4; i += stride) {
        int node = i >> 5;                   // F/4 == 32
        int f4   = (i & 31) * 4;
        float di = dis[node];
        float sl = di * di;
        float4 a = agg[i];
        float4 h = T[i];
        a.x = fmaxf(a.x + h.x * sl + b[f4 + 0], 0.0f);
        a.y = fmaxf(a.y + h.y * sl + b[f4 + 1], 0.0f);
        a.z = fmaxf(a.z + h.z * sl + b[f4 + 2], 0.0f);
        a.w = fmaxf(a.w + h.w * sl + b[f4 + 3], 0.0f);
        agg[i] = a;
    }
}

// ---------------- out = H @ Wp + bp  (N padded 40 -> 48), fp32 WMMA ----------------
// 96 threads = 3 waves; block owns 16 rows x 48 cols
__global__ __launch_bounds__(96) void gcn_gemm48(const float* __restrict__ H,
                                                 const float* __restrict__ Wp,
                                                 const float* __restrict__ bp,
                                                 float* __restrict__ out) {
    __shared__ float As[16 * 132];
    const int t    = threadIdx.x;
    const int row0 = blockIdx.x * 16;

    for (int idx = t; idx < 16 * 128; idx += 96) {
        int r = idx >> 7, c = idx & 127;
        As[r * 132 + c] = H[(size_t)(row0 + r) * F + c];
    }
    __syncthreads();

    const int lane = t & 31;
    const int half = lane >> 4;
    const int m    = lane & 15;
    const int n0   = (t >> 5) * 16;

    v8f acc = {};
    #pragma unroll 8
    for (int k = 0; k < F; k += 4) {
        v2f a, b;
        a.x = As[m * 132 + k + 2 * half];
        a.y = As[m * 132 + k + 2 * half + 1];
        b.x = Wp[(size_t)(k + 2 * half) * NPAD + n0 + m];
        b.y = Wp[(size_t)(k + 2 * half + 1) * NPAD + n0 + m];
        acc = __builtin_amdgcn_wmma_f32_16x16x4_f32(false, a, false, b,
                                                    (short)0, acc, false, false);
    }

    int col = n0 + m;
    if (col < N_CLS) {                       // after all WMMAs (EXEC must be ~0 during WMMA)
        float bias = bp[col];
        #pragma unroll
        for (int r = 0; r < 8; ++r)
            out[(size_t)(row0 + r + 8 * half) * N_CLS + col] = acc[r] + bias;
    }
}

extern "C" void kernel_launch(void* const* d_in, const int* in_sizes, int n_in,
                              void* d_out, int out_size, void* d_ws, size_t ws_size,
                              hipStream_t stream) {
    const float* x    = (const float*)d_in[0];
    const int*   ei   = (const int*)d_in[1];   // [2, E] flat: src row then dst row
    const float* W0   = (const float*)d_in[2];
    const float* b0   = (const float*)d_in[3];
    const float* W1   = (const float*)d_in[4];
    const float* b1   = (const float*)d_in[5];
    const float* W2   = (const float*)d_in[6];
    const float* b2   = (const float*)d_in[7];
    const float* linW = (const float*)d_in[8];
    const float* linb = (const float*)d_in[9];
    float* out = (float*)d_out;

    const size_t NF = (size_t)N_NODES * F;     // 12.8M floats
    float* B0  = (float*)d_ws;                 // GEMM output T
    float* B1  = B0 + NF;                      // agg / layer output (ping-pong partner)
    float* dis = B1 + NF;                      // deg -> dis
    float* Wp  = dis + N_NODES;                // 128x48 padded linW
    float* bp  = Wp + F * NPAD;                // 48

    const int rowBlocks = N_NODES / 16;        // 6250 exactly
    const int n4        = N_NODES * (F / 4);

    // normalization
    gcn_init_deg<<<(N_NODES + 255) / 256, 256, 0, stream>>>(dis);
    gcn_count_deg<<<(N_EDGES + 255) / 256, 256, 0, stream>>>(ei, dis);
    gcn_make_dis<<<(N_NODES + 255) / 256, 256, 0, stream>>>(dis);
    gcn_pad_w<<<(F * NPAD + 255) / 256, 256, 0, stream>>>(linW, linb, Wp, bp);

    const float* Ws[3] = {W0, W1, W2};
    const float* bs[3] = {b0, b1, b2};
    const float* Xin = x;
    for (int l = 0; l < 3; ++l) {
        gcn_gemm128<<<rowBlocks, 256, 0, stream>>>(Xin, Ws[l], B0);
        gcn_zero<<<2048, 256, 0, stream>>>((float4*)B1, n4);
        gcn_scatter<<<4096, 256, 0, stream>>>(ei, B0, dis, B1);
        gcn_finalize<<<2048, 256, 0, stream>>>((float4*)B1, (const float4*)B0, dis, bs[l]);
        Xin = B1;
    }

    gcn_gemm48<<<rowBlocks, 96, 0, stream>>>(B1, Wp, bp, out);
}